// DeformableFeatureAggregation_58858231824547
// MI455X (gfx1250) — compile-verified
//
#include <hip/hip_runtime.h>
#include <math.h>

// ---- problem constants (from reference) ----
#define NA    4096          // num_anchor
#define ED    256           // EMBED_DIMS
#define NG    8             // NUM_GROUPS
#define GD    32            // GROUP_DIMS == wave32 lanes
#define NL    4             // NUM_LEVELS
#define NC    6             // NUM_CAMS
#define NPT   13            // 7 fixed + 6 learnable points
#define NLP   6             // NUM_LEARNABLE_PTS
#define WCOLS 2496          // NG*NC*NL*NPT  (weight logits per anchor)
#define NSAMP 312           // NPT*NC*NL     (samples per anchor)

typedef float v2f __attribute__((ext_vector_type(2)));
typedef float v8f __attribute__((ext_vector_type(8)));

__device__ __constant__ float FIXS[7][3] = {
    {0.f,0.f,0.f},{0.45f,0.f,0.f},{-0.45f,0.f,0.f},
    {0.f,0.45f,0.f},{0.f,-0.45f,0.f},{0.f,0.f,0.45f},{0.f,0.f,-0.45f}};

__device__ __forceinline__ float safe_sigmoid(float x) {
    x = fminf(fmaxf(x, -9.21f), 9.21f);
    return 1.f / (1.f + expf(-x));
}

// ---------------------------------------------------------------------------
// K1: learnable scales + key points + camera projection -> p2d (NA,13,6,2)
// one thread per (anchor, pt); pts padded to 16 per anchor
// ---------------------------------------------------------------------------
__global__ void k1_points(const float* __restrict__ IF, const float* __restrict__ anchor,
                          const float* __restrict__ Wl, const float* __restrict__ bl,
                          const float* __restrict__ proj, const float* __restrict__ wh,
                          float* __restrict__ p2d)
{
    int t  = blockIdx.x * blockDim.x + threadIdx.x;
    int a  = t >> 4;
    int pt = t & 15;
    if (a >= NA || pt >= NPT) return;

    const float* an = anchor + a * 11;
    float sx, sy, sz;
    if (pt < 7) {
        sx = FIXS[pt][0]; sy = FIXS[pt][1]; sz = FIXS[pt][2];
    } else {
        int j = pt - 7;
        float a0 = bl[j*3+0], a1 = bl[j*3+1], a2 = bl[j*3+2];
        const float* ifr = IF + (size_t)a * ED;
        for (int k = 0; k < ED; ++k) {
            float f = ifr[k];
            const float* w = Wl + k * (NLP*3) + j*3;
            a0 = fmaf(f, w[0], a0);
            a1 = fmaf(f, w[1], a1);
            a2 = fmaf(f, w[2], a2);
        }
        sx = safe_sigmoid(a0) - 0.5f;
        sy = safe_sigmoid(a1) - 0.5f;
        sz = safe_sigmoid(a2) - 0.5f;
    }
    // scale by gaussian scales
    float kx = sx * an[3], ky = sy * an[4], kz = sz * an[5];
    // rotation (R^T applied)
    float qw = an[6], qx = an[7], qy = an[8], qz = an[9];
    float qn = rsqrtf(qw*qw + qx*qx + qy*qy + qz*qz);
    qw *= qn; qx *= qn; qy *= qn; qz *= qn;
    float R00 = 1.f-2.f*(qy*qy+qz*qz), R01 = 2.f*(qx*qy-qw*qz), R02 = 2.f*(qx*qz+qw*qy);
    float R10 = 2.f*(qx*qy+qw*qz), R11 = 1.f-2.f*(qx*qx+qz*qz), R12 = 2.f*(qy*qz-qw*qx);
    float R20 = 2.f*(qx*qz-qw*qy), R21 = 2.f*(qy*qz+qw*qx), R22 = 1.f-2.f*(qx*qx+qy*qy);
    // out_i = sum_j R[j][i] * kp[j]  (rot_T einsum)
    float px = R00*kx + R10*ky + R20*kz + an[0];
    float py = R01*kx + R11*ky + R21*kz + an[1];
    float pz = R02*kx + R12*ky + R22*kz + an[2];

    for (int c = 0; c < NC; ++c) {
        const float* P = proj + c * 16;
        float u = P[0]*px + P[1]*py + P[2]*pz  + P[3];
        float v = P[4]*px + P[5]*py + P[6]*pz  + P[7];
        float w = P[8]*px + P[9]*py + P[10]*pz + P[11];
        w = fmaxf(w, 1e-5f);
        float ux = (u / w) / wh[c*2+0];
        float uy = (v / w) / wh[c*2+1];
        ux = fminf(fmaxf(ux, 0.f), 0.9999f);
        uy = fminf(fmaxf(uy, 0.f), 0.9999f);
        float* o = p2d + (((size_t)(a*NPT + pt) * NC + c) * 2);
        o[0] = ux; o[1] = uy;
    }
}

// ---------------------------------------------------------------------------
// K2/K4: f32 WMMA GEMM  out(M=4096,N) = A(M,256) @ B(256,N) + bias [+ addend]
// block = 128 threads (4 waves), each wave owns one 16x16 tile, grid = (N/64, M/16)
// A tile staged into LDS via GLOBAL_LOAD_ASYNC_TO_LDS_B128 (CDNA5 direct
// memory->LDS path, ASYNCcnt-tracked; bypasses the VGPR round trip).
// LDS pitch 260 floats (1040B = 65*16B, still 16B aligned) kills the
// 64-bank conflict at stride 256.
// ---------------------------------------------------------------------------
template <bool HASADD>
__global__ void wmma_gemm_f32(const float* __restrict__ A, const float* __restrict__ B,
                              const float* __restrict__ bias, const float* __restrict__ addend,
                              float* __restrict__ out, int N)
{
    __shared__ __align__(16) float sA[16 * 260];
    int tid = threadIdx.x;
    int m0  = blockIdx.y * 16;

    // 16 rows x 256 floats = 1024 x b128 chunks; 128 threads x 8 iters
    #pragma unroll
    for (int it = 0; it < 8; ++it) {
        int idx = tid + it * 128;           // b128 chunk id
        int r   = idx >> 6;                 // row (0..15)
        int kc  = idx & 63;                 // 4-float chunk within row
        const float* gp = A + (size_t)(m0 + r) * ED + kc * 4;
        unsigned ldsOff = (unsigned)(unsigned long long)(uintptr_t)&sA[r * 260 + kc * 4];
        asm volatile("global_load_async_to_lds_b128 %0, %1, off"
                     :: "v"(ldsOff), "v"(gp) : "memory");
    }
    asm volatile("s_wait_asynccnt 0" ::: "memory");
    __syncthreads();

    int wave = tid >> 5, lane = tid & 31;
    int half = lane >> 4, mrow = lane & 15;
    int n0   = (blockIdx.x * 4 + wave) * 16;
    int col  = n0 + mrow;

    v8f c = {0.f,0.f,0.f,0.f,0.f,0.f,0.f,0.f};
    const float* arow = sA + mrow * 260 + 2 * half;   // K offset 2*half for this lane
    #pragma unroll 8
    for (int kk = 0; kk < 64; ++kk) {
        int k0 = kk * 4 + 2 * half;
        v2f a, b;
        a.x = arow[kk * 4];
        a.y = arow[kk * 4 + 1];
        b.x = B[(size_t)k0 * N + col];
        b.y = B[(size_t)(k0 + 1) * N + col];
        c = __builtin_amdgcn_wmma_f32_16x16x4_f32(
                /*neg_a=*/false, a, /*neg_b=*/false, b,
                /*c_mod=*/(short)0, c, /*reuse_a=*/false, /*reuse_b=*/false);
    }

    float bcol = bias[col];
    #pragma unroll
    for (int v = 0; v < 8; ++v) {
        int m = m0 + v + 8 * half;          // C/D layout: M = vgpr + 8*(lane>=16)
        float r = c[v] + bcol;
        if (HASADD) r += addend[(size_t)m * N + col];
        out[(size_t)m * N + col] = r;
    }
}

// ---------------------------------------------------------------------------
// K2b: in-place softmax over the 312 samples per (anchor, group), with
// reorder (cam,level,pt,g) -> (pt,cam,level,g). One block per anchor,
// wave w handles group w (columns are wave-private, so no extra syncs).
// ---------------------------------------------------------------------------
__global__ void k2b_softmax(float* __restrict__ wts)
{
    __shared__ float row[WCOLS];
    int a = blockIdx.x, tid = threadIdx.x;
    float* g_row = wts + (size_t)a * WCOLS;
    for (int i = tid; i < WCOLS; i += 256) row[i] = g_row[i];
    __syncthreads();

    int g = tid >> 5, lane = tid & 31;
    float m = -1e30f;
    for (int k = lane; k < NSAMP; k += 32) m = fmaxf(m, row[k * NG + g]);
    for (int o = 16; o; o >>= 1) m = fmaxf(m, __shfl_xor(m, o, 32));

    float s = 0.f;
    for (int k = lane; k < NSAMP; k += 32) {
        float e = expf(row[k * NG + g] - m);
        row[k * NG + g] = e;
        s += e;
    }
    for (int o = 16; o; o >>= 1) s += __shfl_xor(s, o, 32);
    float inv = 1.f / s;

    for (int k = lane; k < NSAMP; k += 32) {
        int cam = k / (NL * NPT);         // k = (cam*4+level)*13+pt
        int rem = k % (NL * NPT);
        int level = rem / NPT;
        int pt    = rem % NPT;
        int dst = ((pt * NC + cam) * NL + level) * NG + g;   // aggregation layout
        g_row[dst] = row[k * NG + g] * inv;
    }
}

// ---------------------------------------------------------------------------
// K3: bilinear gather + weighted aggregate. One block per anchor, one thread
// per channel (group = tid>>5 is wave-uniform -> all LDS reads broadcast,
// all bounds branches uniform). Feature pyramids (~92MB) are L2-resident
// (192MB L2), so the ~1.3e9 gathers stay on-chip.
// ---------------------------------------------------------------------------
__global__ void k3_aggregate(const float* __restrict__ p2d, const float* __restrict__ wts,
                             const float* __restrict__ f0, const float* __restrict__ f1,
                             const float* __restrict__ f2, const float* __restrict__ f3,
                             float* __restrict__ feats)
{
    __shared__ float s_w[WCOLS];
    __shared__ float s_wx[NSAMP], s_wy[NSAMP];
    __shared__ int   s_x0[NSAMP], s_y0[NSAMP];

    int a = blockIdx.x, tid = threadIdx.x;
    for (int i = tid; i < WCOLS; i += 256) s_w[i] = wts[(size_t)a * WCOLS + i];
    for (int s = tid; s < NSAMP; s += 256) {
        // s = ((pt*6 + cam)*4 + level)  -> weight index is simply s*8+g
        int pt  = s / (NC * NL);
        int cam = (s >> 2) % NC;
        int l   = s & 3;
        int Wl  = (l == 0) ? 176 : (l == 1) ? 88 : (l == 2) ? 44 : 22;
        int Hl  = (l == 0) ?  64 : (l == 1) ? 32 : (l == 2) ? 16 :  8;
        const float* o = p2d + (((size_t)(a*NPT + pt) * NC + cam) * 2);
        float x = o[0] * (float)Wl - 0.5f;
        float y = o[1] * (float)Hl - 0.5f;
        float xf = floorf(x), yf = floorf(y);
        s_x0[s] = (int)xf; s_y0[s] = (int)yf;
        s_wx[s] = x - xf;  s_wy[s] = y - yf;
    }
    __syncthreads();

    int d = tid;
    int g = tid >> 5;          // == wave id; LDS weight reads are broadcasts
    float acc = 0.f;

    for (int s = 0; s < NSAMP; ++s) {
        int l   = s & 3;
        int cam = (s >> 2) % NC;
        int W   = (l == 0) ? 176 : (l == 1) ? 88 : (l == 2) ? 44 : 22;
        int H   = (l == 0) ?  64 : (l == 1) ? 32 : (l == 2) ? 16 :  8;
        const float* fb = (l == 0) ? f0 : (l == 1) ? f1 : (l == 2) ? f2 : f3;
        int HW = W * H;
        const float* base = fb + (size_t)(cam * ED + d) * HW;

        float w  = s_w[s * NG + g];
        int x0 = s_x0[s], y0 = s_y0[s];
        float wx = s_wx[s], wy = s_wy[s];
        bool xv0 = ((unsigned)x0       < (unsigned)W);
        bool xv1 = ((unsigned)(x0 + 1) < (unsigned)W);
        bool yv0 = ((unsigned)y0       < (unsigned)H);
        bool yv1 = ((unsigned)(y0 + 1) < (unsigned)H);

        float v00 = (xv0 && yv0) ? base[y0 * W + x0]           : 0.f;
        float v01 = (xv1 && yv0) ? base[y0 * W + x0 + 1]       : 0.f;
        float v10 = (xv0 && yv1) ? base[(y0 + 1) * W + x0]     : 0.f;
        float v11 = (xv1 && yv1) ? base[(y0 + 1) * W + x0 + 1] : 0.f;

        float top = v00 * (1.f - wx) + v01 * wx;
        float bot = v10 * (1.f - wx) + v11 * wx;
        acc = fmaf(w, top * (1.f - wy) + bot * wy, acc);
    }
    feats[(size_t)a * ED + d] = acc;
}

// ---------------------------------------------------------------------------
extern "C" void kernel_launch(void* const* d_in, const int* in_sizes, int n_in,
                              void* d_out, int out_size, void* d_ws, size_t ws_size,
                              hipStream_t stream) {
    (void)in_sizes; (void)n_in; (void)out_size; (void)ws_size;
    // d_in[0] = pts3d (unused)
    const float* IF     = (const float*)d_in[1];
    const float* anchor = (const float*)d_in[2];
    const float* f0     = (const float*)d_in[3];
    const float* f1     = (const float*)d_in[4];
    const float* f2     = (const float*)d_in[5];
    const float* f3     = (const float*)d_in[6];
    const float* proj   = (const float*)d_in[7];
    const float* wh     = (const float*)d_in[8];
    const float* Wl     = (const float*)d_in[9];
    const float* bl     = (const float*)d_in[10];
    const float* Wwts   = (const float*)d_in[11];
    const float* bwts   = (const float*)d_in[12];
    const float* Wout   = (const float*)d_in[13];
    const float* bout   = (const float*)d_in[14];
    float* out = (float*)d_out;

    float* ws    = (float*)d_ws;
    float* p2d   = ws;                                    // NA*13*6*2   = 2.6 MB
    float* wlog  = p2d  + (size_t)NA * NPT * NC * 2;      // NA*2496     = 40.9 MB
    float* feats = wlog + (size_t)NA * WCOLS;             // NA*256      = 4 MB

    k1_points           <<<dim3(NA * 16 / 256), 256, 0, stream>>>(IF, anchor, Wl, bl, proj, wh, p2d);
    wmma_gemm_f32<false><<<dim3(WCOLS / 64, NA / 16), 128, 0, stream>>>(IF, Wwts, bwts, nullptr, wlog, WCOLS);
    k2b_softmax         <<<dim3(NA), 256, 0, stream>>>(wlog);
    k3_aggregate        <<<dim3(NA), 256, 0, stream>>>(p2d, wlog, f0, f1, f2, f3, feats);
    wmma_gemm_f32<true> <<<dim3(ED / 64, NA / 16), 128, 0, stream>>>(feats, Wout, bout, IF, out, ED);
}